// Patch3DGPNNDirectLoss_40810779246800
// MI455X (gfx1250) — compile-verified
//
#include <hip/hip_runtime.h>
#include <hip/hip_bf16.h>

// Patch3D GPNN direct loss for x,y:(1,3,20,50,50), P=TP=7, S=ST=1.
// d_out=14, h_out=w_out=44 -> 1936 locations, patch length C=1029.
//
// Stage 1: one wave32 per spatial location computes the full (padded 16x16)
//          fp32 Gram matrix of x-patches vs y-patches with
//          V_WMMA_F32_16X16X4_F32, accumulates patch norms alongside, then
//          does an in-register argmin (distance = xsq + ysq - 2*dot) with
//          first-index tie-break. Writes nns[1936][14] to workspace.
// Stage 2: one thread per voxel gathers the <=343 contributing y values via
//          nns, forms folded/weight and |x - y2x|*weight, block-reduces.
// Stage 3: deterministic single-block final reduction -> loss scalar.

typedef float v2f __attribute__((ext_vector_type(2)));
typedef float v8f __attribute__((ext_vector_type(8)));

#define NLOC   1936   // 44*44
#define DOUT   14
#define HOUT   44
#define CH_STR 50000  // 20*50*50 elements per channel
#define D_STR  2500   // 50*50

__global__ __launch_bounds__(256)
void gpnn_nn_kernel(const float* __restrict__ x,
                    const float* __restrict__ y,
                    int* __restrict__ nns) {
  const int wid  = blockIdx.x * 8 + (threadIdx.x >> 5);   // one wave = one location
  const int lane = threadIdx.x & 31;
  if (wid >= NLOC) return;                                 // uniform per wave

  const int ho = wid / HOUT;
  const int wo = wid % HOUT;

  const int   col   = lane & 15;            // A-row index n / B-col index m
  const int   hi    = lane >> 4;            // lane-half selects K pair {0,1} vs {2,3}
  const int   kOff  = hi << 1;
  const int   rowv  = (col < 14) ? col : 0; // clamp padded rows for addressing
  const float rmask = (col < 14) ? 1.0f : 0.0f;

  v8f acc = {0.f,0.f,0.f,0.f,0.f,0.f,0.f,0.f};
  float sqx = 0.f, sqy = 0.f;

  // K space laid out as (c, kd, kh, kw) with kw padded 7->8 so each 4-wide
  // K chunk is uniform in (c,kd,kh): zero divisions in the hot loop.
  for (int cc = 0; cc < 3; ++cc) {
    for (int kdd = 0; kdd < 7; ++kdd) {
      const int dep = cc * 20 + kdd + rowv;
      for (int khh = 0; khh < 7; ++khh) {
        const int base = (dep * 50 + ho + khh) * 50 + wo;
#pragma unroll
        for (int kwb = 0; kwb < 8; kwb += 4) {
          const int k0 = kwb + kOff;                 // k0 in [0,6]
          const float m1 = (k0 + 1 < 7) ? rmask : 0.0f;
          const int i0 = base + k0;
          const int i1 = base + ((k0 + 1 < 7) ? (k0 + 1) : 0);
          const float a0 = rmask * x[i0];
          const float a1 = m1    * x[i1];
          const float b0 = rmask * y[i0];
          const float b1 = m1    * y[i1];
          v2f A = {a0, a1};
          v2f B = {b0, b1};
          acc = __builtin_amdgcn_wmma_f32_16x16x4_f32(
              false, A, false, B, (short)0, acc, false, false);
          sqx += a0 * a0 + a1 * a1;
          sqy += b0 * b0 + b1 * b1;
        }
      }
    }
  }

  // Combine the two K-halves of the squared norms: lane L holds rows (L&15).
  const float xsq = sqx + __shfl_xor(sqx, 16, 32);  // xsq[col], in lanes col and col+16
  const float ysq = sqy + __shfl_xor(sqy, 16, 32);  // ysq[col]

  // Per-row argmin over m (columns are striped across the 16 lanes of a half).
#pragma unroll
  for (int v = 0; v < 8; ++v) {
    const int M = v + hi * 8;                         // Gram row of acc[v]
    const float xr = __shfl(xsq, M, 32);              // lane M (<16) holds xsq[M]
    const float dist = xr + ysq - 2.0f * acc[v];
    float bv = (col < 14) ? dist : 3.0e38f;           // mask padded columns
    int   bi = col;
#pragma unroll
    for (int m = 1; m < 16; m <<= 1) {                // stays within lane-half
      const float ov = __shfl_xor(bv, m, 32);
      const int   oi = __shfl_xor(bi, m, 32);
      if (ov < bv || (ov == bv && oi < bi)) { bv = ov; bi = oi; }
    }
    if ((col == v) && (M < 14)) nns[wid * DOUT + M] = bi;
  }
}

__global__ __launch_bounds__(256)
void gpnn_fold_loss_kernel(const float* __restrict__ x,
                           const float* __restrict__ y,
                           const int* __restrict__ nns,
                           float* __restrict__ partials) {
  __shared__ float red[256];
  const int tid = blockIdx.x * 256 + threadIdx.x;
  float contrib = 0.0f;
  if (tid < 50000) {
    const int d  = tid / D_STR;
    const int hw = tid % D_STR;
    const int h  = hw / 50;
    const int w  = hw % 50;
    const int ho0 = (h - 6 > 0)  ? h - 6 : 0;
    const int ho1 = (h < 43)     ? h     : 43;
    const int wo0 = (w - 6 > 0)  ? w - 6 : 0;
    const int wo1 = (w < 43)     ? w     : 43;
    const int n0  = (d - 6 > 0)  ? d - 6 : 0;
    const int n1  = (d < 13)     ? d     : 13;

    float a0 = 0.f, a1 = 0.f, a2 = 0.f;
    int cnt = 0;
    for (int ho = ho0; ho <= ho1; ++ho) {
      for (int wo = wo0; wo <= wo1; ++wo) {
        const int* nrow = nns + (ho * HOUT + wo) * DOUT;
        for (int n = n0; n <= n1; ++n) {
          const int m   = nrow[n];
          const int off = (m + d - n) * D_STR + hw;  // y depth = nns + kd
          a0 += y[off];
          a1 += y[off + CH_STR];
          a2 += y[off + 2 * CH_STR];
          ++cnt;
        }
      }
    }
    const float wgt = (float)cnt;
    const float inv = 1.0f / wgt;
    const int xo = d * D_STR + hw;
    contrib = (fabsf(x[xo]              - a0 * inv) +
               fabsf(x[xo + CH_STR]     - a1 * inv) +
               fabsf(x[xo + 2 * CH_STR] - a2 * inv)) * wgt;
  }
  red[threadIdx.x] = contrib;
  __syncthreads();
  for (int s = 128; s > 0; s >>= 1) {
    if (threadIdx.x < s) red[threadIdx.x] += red[threadIdx.x + s];
    __syncthreads();
  }
  if (threadIdx.x == 0) partials[blockIdx.x] = red[0];
}

__global__ __launch_bounds__(256)
void gpnn_final_reduce(const float* __restrict__ partials, int np,
                       float* __restrict__ out) {
  __shared__ float red[256];
  float s = 0.0f;
  for (int i = threadIdx.x; i < np; i += 256) s += partials[i];
  red[threadIdx.x] = s;
  __syncthreads();
  for (int k = 128; k > 0; k >>= 1) {
    if (threadIdx.x < k) red[threadIdx.x] += red[threadIdx.x + k];
    __syncthreads();
  }
  if (threadIdx.x == 0) out[0] = red[0] / 150000.0f;  // mean over 3*20*50*50
}

extern "C" void kernel_launch(void* const* d_in, const int* in_sizes, int n_in,
                              void* d_out, int out_size, void* d_ws, size_t ws_size,
                              hipStream_t stream) {
  const float* x = (const float*)d_in[0];
  const float* y = (const float*)d_in[1];
  float* out = (float*)d_out;

  int* nns = (int*)d_ws;
  const size_t nns_bytes = (size_t)NLOC * DOUT * sizeof(int);
  float* partials = (float*)((char*)d_ws + ((nns_bytes + 255) & ~(size_t)255));

  // 242 blocks * 8 waves = 1936 waves = one per spatial location.
  gpnn_nn_kernel<<<242, 256, 0, stream>>>(x, y, nns);
  // 50000 voxels, 196 blocks of 256.
  gpnn_fold_loss_kernel<<<196, 256, 0, stream>>>(x, y, nns, partials);
  gpnn_final_reduce<<<1, 256, 0, stream>>>(partials, 196, out);
}